// IGR_38182259261717
// MI455X (gfx1250) — compile-verified
//
#include <hip/hip_runtime.h>
#include <math.h>

typedef __attribute__((ext_vector_type(2))) float v2f;
typedef __attribute__((ext_vector_type(8))) float v8f;

#define D 128
#define EPS 1e-5f

__device__ __forceinline__ float sigmoidf(float x) {
    return 1.0f / (1.0f + __expf(-x));
}

// ---------------------------------------------------------------------------
// Instance norm over rows of length 128: one wave per row, float4 per lane.
// ---------------------------------------------------------------------------
__global__ __launch_bounds__(256) void k_inorm(const float* __restrict__ x,
                                               float* __restrict__ y, int nrows) {
    int wave = (blockIdx.x * blockDim.x + threadIdx.x) >> 5;
    int lane = threadIdx.x & 31;
    if (wave >= nrows) return;
    const float4* xr = (const float4*)(x + (size_t)wave * D);
    float4 v = xr[lane];
    float s  = v.x + v.y + v.z + v.w;
    float ss = v.x * v.x + v.y * v.y + v.z * v.z + v.w * v.w;
    #pragma unroll
    for (int m = 16; m >= 1; m >>= 1) {
        s  += __shfl_xor(s,  m, 32);
        ss += __shfl_xor(ss, m, 32);
    }
    float mean = s * (1.0f / D);
    float var  = ss * (1.0f / D) - mean * mean;
    float inv  = rsqrtf(var + EPS);
    float4 o;
    o.x = (v.x - mean) * inv; o.y = (v.y - mean) * inv;
    o.z = (v.z - mean) * inv; o.w = (v.w - mean) * inv;
    ((float4*)(y + (size_t)wave * D))[lane] = o;
}

// ---------------------------------------------------------------------------
// Y = inorm(X @ W^T + b).  X: nrows x 128, W: 128 x 128 (row major), K = 128.
// Block = 256 thr (8 waves); block does 16 rows x 128 cols; wave w -> cols
// [16w,16w+16).  WMMA f32 16x16x4, A staged in LDS, epilogue = fused inorm.
// ---------------------------------------------------------------------------
__global__ __launch_bounds__(256) void k_gemm128_inorm(
    const float* __restrict__ X, const float* __restrict__ W,
    const float* __restrict__ bias, float* __restrict__ Y, int nrows) {
    __shared__ float Xs[16 * 128];
    __shared__ float Ys[16 * 128];
    __shared__ float red0[16][16];
    __shared__ float red1[16][16];
    __shared__ float mean_s[16], inv_s[16];

    int t = threadIdx.x;
    int rowBase = blockIdx.x * 16;
    #pragma unroll
    for (int i = 0; i < 8; i++) {
        int idx = t + 256 * i;                    // 0..2047
        int r = idx >> 7, c = idx & 127;
        int rr = rowBase + r; if (rr >= nrows) rr = nrows - 1;
        Xs[idx] = X[(size_t)rr * D + c];
    }
    __syncthreads();

    int lane = t & 31, wave = t >> 5;
    int half = lane >> 4, l16 = lane & 15;
    int nCol = wave * 16 + l16;                   // output column (W row)
    v8f acc = {};
    #pragma unroll 4
    for (int kb = 0; kb < 128; kb += 4) {
        int k = kb + 2 * half;
        v2f a = *(const v2f*)&Xs[l16 * 128 + k];
        v2f b = *(const v2f*)&W[(size_t)nCol * 128 + k];
        acc = __builtin_amdgcn_wmma_f32_16x16x4_f32(false, a, false, b,
                                                    (short)0, acc, false, false);
    }
    float bn = bias[nCol];
    #pragma unroll
    for (int r = 0; r < 8; r++) {
        int m = r + 8 * half;                     // row within tile
        Ys[m * 128 + nCol] = acc[r] + bn;
    }
    __syncthreads();

    // per-row mean/var: 16 threads per row, 8 elems each
    int row = t >> 4, sub = t & 15;
    float s = 0.f, ss = 0.f;
    #pragma unroll
    for (int j = 0; j < 8; j++) {
        float v = Ys[row * 128 + sub + 16 * j];
        s += v; ss += v * v;
    }
    red0[row][sub] = s; red1[row][sub] = ss;
    __syncthreads();
    for (int off = 8; off >= 1; off >>= 1) {
        if (sub < off) {
            red0[row][sub] += red0[row][sub + off];
            red1[row][sub] += red1[row][sub + off];
        }
        __syncthreads();
    }
    if (sub == 0) {
        float mean = red0[row][0] * (1.0f / D);
        float var  = red1[row][0] * (1.0f / D) - mean * mean;
        mean_s[row] = mean;
        inv_s[row]  = rsqrtf(var + EPS);
    }
    __syncthreads();
    #pragma unroll
    for (int i = 0; i < 8; i++) {
        int idx = t + 256 * i;
        int r = idx >> 7, c = idx & 127;
        if (rowBase + r < nrows)
            Y[(size_t)(rowBase + r) * D + c] = (Ys[idx] - mean_s[r]) * inv_s[r];
    }
}

// ---------------------------------------------------------------------------
// Y = sigmoid(concat(E1,X2) @ W^T + b).  K = 256, W: 128 x 256.
// ---------------------------------------------------------------------------
__global__ __launch_bounds__(256) void k_gemm256_concat_sig(
    const float* __restrict__ E1, const float* __restrict__ X2,
    const float* __restrict__ W, const float* __restrict__ bias,
    float* __restrict__ Y, int nrows) {
    __shared__ float Xs[16 * 256];
    int t = threadIdx.x;
    int rowBase = blockIdx.x * 16;
    #pragma unroll
    for (int i = 0; i < 16; i++) {
        int idx = t + 256 * i;                    // 0..4095
        int r = idx >> 8, c = idx & 255;
        int rr = rowBase + r; if (rr >= nrows) rr = nrows - 1;
        Xs[idx] = (c < 128) ? E1[(size_t)rr * D + c]
                            : X2[(size_t)rr * D + (c - 128)];
    }
    __syncthreads();

    int lane = t & 31, wave = t >> 5;
    int half = lane >> 4, l16 = lane & 15;
    int nCol = wave * 16 + l16;
    v8f acc = {};
    #pragma unroll 4
    for (int kb = 0; kb < 256; kb += 4) {
        int k = kb + 2 * half;
        v2f a = *(const v2f*)&Xs[l16 * 256 + k];
        v2f b = *(const v2f*)&W[(size_t)nCol * 256 + k];
        acc = __builtin_amdgcn_wmma_f32_16x16x4_f32(false, a, false, b,
                                                    (short)0, acc, false, false);
    }
    float bn = bias[nCol];
    #pragma unroll
    for (int r = 0; r < 8; r++) {
        int m = r + 8 * half;
        if (rowBase + m < nrows)
            Y[(size_t)(rowBase + m) * D + nCol] = sigmoidf(acc[r] + bn);
    }
}

// ---------------------------------------------------------------------------
// Y = sigmoid(relu(X @ W^T + b)).  K = 128 (social layer).
// ---------------------------------------------------------------------------
__global__ __launch_bounds__(256) void k_gemm128_relu_sig(
    const float* __restrict__ X, const float* __restrict__ W,
    const float* __restrict__ bias, float* __restrict__ Y, int nrows) {
    __shared__ float Xs[16 * 128];
    int t = threadIdx.x;
    int rowBase = blockIdx.x * 16;
    #pragma unroll
    for (int i = 0; i < 8; i++) {
        int idx = t + 256 * i;
        int r = idx >> 7, c = idx & 127;
        int rr = rowBase + r; if (rr >= nrows) rr = nrows - 1;
        Xs[idx] = X[(size_t)rr * D + c];
    }
    __syncthreads();

    int lane = t & 31, wave = t >> 5;
    int half = lane >> 4, l16 = lane & 15;
    int nCol = wave * 16 + l16;
    v8f acc = {};
    #pragma unroll 4
    for (int kb = 0; kb < 128; kb += 4) {
        int k = kb + 2 * half;
        v2f a = *(const v2f*)&Xs[l16 * 128 + k];
        v2f b = *(const v2f*)&W[(size_t)nCol * 128 + k];
        acc = __builtin_amdgcn_wmma_f32_16x16x4_f32(false, a, false, b,
                                                    (short)0, acc, false, false);
    }
    float bn = bias[nCol];
    #pragma unroll
    for (int r = 0; r < 8; r++) {
        int m = r + 8 * half;
        if (rowBase + m < nrows) {
            float v = acc[r] + bn;
            v = v > 0.f ? v : 0.f;
            Y[(size_t)(rowBase + m) * D + nCol] = sigmoidf(v);
        }
    }
}

// ---------------------------------------------------------------------------
// SpMM with SORTED COO rows: one block per output row; binary-search the
// contiguous edge range, reduce, write (deterministic, no atomics).
// ---------------------------------------------------------------------------
__global__ __launch_bounds__(128) void k_spmm_row(
    const int* __restrict__ rows, const int* __restrict__ cols,
    const float* __restrict__ vals, const float* __restrict__ X,
    float* __restrict__ Y, int nnz, int applySig) {
    int r = blockIdx.x;
    __shared__ int bnd[2];
    if (threadIdx.x == 0) {
        int lo = 0, hi = nnz;
        while (lo < hi) { int mid = (lo + hi) >> 1; if (rows[mid] < r) lo = mid + 1; else hi = mid; }
        bnd[0] = lo;
        hi = nnz;
        while (lo < hi) { int mid = (lo + hi) >> 1; if (rows[mid] < r + 1) lo = mid + 1; else hi = mid; }
        bnd[1] = lo;
    }
    __syncthreads();
    int e0 = bnd[0], e1 = bnd[1];
    int d = threadIdx.x;
    float acc = 0.f;
    for (int e = e0; e < e1; e++)
        acc += vals[e] * X[(size_t)cols[e] * D + d];   // uniform col/val, gathered row (L2-resident)
    if (applySig) acc = sigmoidf(acc);
    Y[(size_t)r * D + d] = acc;
}

// ---------------------------------------------------------------------------
// final_group = sigmoid((gfi + first + second) * 0.5)
// ---------------------------------------------------------------------------
__global__ __launch_bounds__(256) void k_combine(
    const float* __restrict__ a, const float* __restrict__ b,
    const float* __restrict__ c, float* __restrict__ y, int n) {
    int i = blockIdx.x * blockDim.x + threadIdx.x;
    if (i < n) y[i] = sigmoidf((a[i] + b[i] + c[i]) * 0.5f);
}

// ---------------------------------------------------------------------------
// Scoring: wave per pair; gamma = <ge,ie>; gamma2 = <ne,ge> + <ne,ie>.
// ---------------------------------------------------------------------------
__global__ __launch_bounds__(256) void k_score(
    const int* __restrict__ gid, const int* __restrict__ iid,
    const float* __restrict__ FG, const float* __restrict__ FI,
    const float* __restrict__ AN, float* __restrict__ out, int B) {
    int w = (blockIdx.x * blockDim.x + threadIdx.x) >> 5;
    int lane = threadIdx.x & 31;
    if (w >= B) return;
    int g = gid[w], it = iid[w];
    float4 a = ((const float4*)(FG + (size_t)g * D))[lane];
    float4 b = ((const float4*)(FI + (size_t)it * D))[lane];
    float4 c = ((const float4*)(AN + (size_t)g * D))[lane];
    float d0 = a.x * b.x + a.y * b.y + a.z * b.z + a.w * b.w;
    float d1 = c.x * a.x + c.y * a.y + c.z * a.z + c.w * a.w;
    float d2 = c.x * b.x + c.y * b.y + c.z * b.z + c.w * b.w;
    #pragma unroll
    for (int m = 16; m >= 1; m >>= 1) {
        d0 += __shfl_xor(d0, m, 32);
        d1 += __shfl_xor(d1, m, 32);
        d2 += __shfl_xor(d2, m, 32);
    }
    if (lane == 0) {
        out[w]     = d0;
        out[B + w] = d1 + d2;
    }
}

// ---------------------------------------------------------------------------
extern "C" void kernel_launch(void* const* d_in, const int* in_sizes, int n_in,
                              void* d_out, int out_size, void* d_ws, size_t ws_size,
                              hipStream_t stream) {
    const int*   group_ids   = (const int*)  d_in[0];
    const int*   item_ids    = (const int*)  d_in[1];
    const int*   gi_rows     = (const int*)  d_in[2];
    const int*   gi_cols     = (const int*)  d_in[3];
    const float* gi_vals     = (const float*)d_in[4];
    const int*   gg_rows     = (const int*)  d_in[5];
    const int*   gg_cols     = (const int*)  d_in[6];
    const float* gg_vals     = (const float*)d_in[7];
    const float* g_feat      = (const float*)d_in[8];
    const float* i_feat      = (const float*)d_in[9];
    const float* emb_group   = (const float*)d_in[10];
    const float* emb_item    = (const float*)d_in[11];
    const float* W_w         = (const float*)d_in[12];
    const float* W_b         = (const float*)d_in[13];
    const float* red_w       = (const float*)d_in[14];
    const float* red_b       = (const float*)d_in[15];
    const float* item_fus_w  = (const float*)d_in[16];
    const float* item_fus_b  = (const float*)d_in[17];
    const float* group_fus_w = (const float*)d_in[18];
    const float* group_fus_b = (const float*)d_in[19];

    const int B  = in_sizes[0];
    const int E  = in_sizes[2];
    const int NG = in_sizes[8] / D;
    const int NI = in_sizes[9] / D;
    const size_t szNG = (size_t)NG * D;
    const size_t szNI = (size_t)NI * D;

    // workspace layout (slot reuse keeps total at 2*NI + 5*NG rows)
    float* w   = (float*)d_ws;
    float* tI  = w;                  // inorm(i_feat)  -> final_item
    float* i2  = tI + szNI;          // i2
    float* tG  = i2 + szNI;          // inorm(g_feat) -> fus_group -> all_neighbors
    float* g2  = tG + szNG;          // g2 -> s1 -> s2
    float* fst = g2 + szNG;          // first
    float* snd = fst + szNG;         // second
    float* gfi = snd + szNG;         // g_from_items -> final_group
    (void)ws_size; (void)n_in; (void)out_size;

    dim3 blk256(256), blk128(128);
    int gI16 = (NI + 15) / 16, gG16 = (NG + 15) / 16;
    int gI8  = (NI + 7) / 8,   gG8  = (NG + 7) / 8;

    // ---- item path ----
    k_inorm<<<gI8, blk256, 0, stream>>>(i_feat, tI, NI);
    k_gemm128_inorm<<<gI16, blk256, 0, stream>>>(tI, red_w, red_b, i2, NI);
    k_gemm256_concat_sig<<<gI16, blk256, 0, stream>>>(emb_item, i2, item_fus_w,
                                                      item_fus_b, tI, NI); // tI := final_item
    // ---- group path ----
    k_inorm<<<gG8, blk256, 0, stream>>>(g_feat, tG, NG);
    k_gemm128_inorm<<<gG16, blk256, 0, stream>>>(tG, red_w, red_b, g2, NG);
    k_gemm256_concat_sig<<<gG16, blk256, 0, stream>>>(emb_group, g2, group_fus_w,
                                                      group_fus_b, tG, NG); // tG := fus_group
    // ---- g_from_items = sigmoid(spmm(gi, final_item)) ----
    k_spmm_row<<<NG, blk128, 0, stream>>>(gi_rows, gi_cols, gi_vals, tI, gfi, E, 1);
    // ---- first = sigmoid(relu(W(spmm(gg, fus_group)))) ----
    k_spmm_row<<<NG, blk128, 0, stream>>>(gg_rows, gg_cols, gg_vals, tG, g2, E, 0);
    k_gemm128_relu_sig<<<gG16, blk256, 0, stream>>>(g2, W_w, W_b, fst, NG);
    // ---- second = sigmoid(relu(W(spmm(gg, first)))) ----
    k_spmm_row<<<NG, blk128, 0, stream>>>(gg_rows, gg_cols, gg_vals, fst, g2, E, 0);
    k_gemm128_relu_sig<<<gG16, blk256, 0, stream>>>(g2, W_w, W_b, snd, NG);
    // ---- final_group = sigmoid((gfi + first + second)/2) ----
    k_combine<<<(int)((szNG + 255) / 256), blk256, 0, stream>>>(gfi, fst, snd, gfi, (int)szNG);
    // ---- all_neighbors = spmm(gg, final_group) ----
    k_spmm_row<<<NG, blk128, 0, stream>>>(gg_rows, gg_cols, gg_vals, gfi, tG, E, 0);
    // ---- scoring ----
    k_score<<<(B + 7) / 8, blk256, 0, stream>>>(group_ids, item_ids, gfi, tI, tG,
                                                (float*)d_out, B);
}